// EnGMF_48155173322957
// MI455X (gfx1250) — compile-verified
//
#include <hip/hip_runtime.h>
#include <math.h>

typedef __attribute__((ext_vector_type(2))) float v2f;
typedef __attribute__((ext_vector_type(8))) float v8f;

#define NB_GRAM 512
#define NB_RED  512
#define CHUNK   512
#define TILE_P  1024                     // particles per LDS tile
#define TILE_F  (TILE_P * 6)             // floats per tile (24 KB)
#define LOG2PI_F 1.8378770664093453f

// ---------------------------------------------------------------------------
// Stage 1: Gram = X^T X (6x6 padded to 16x16) + column sums, via f32 WMMA.
// Full tiles: async global->LDS staging (coalesced b128 per lane), then the
// per-lane scatter (lane = state dim, 4 particles / WMMA K=4) reads from LDS.
// Tail: branchless clamped global loads.
// A (16x4): lanes 0-15 -> v0=K0,v1=K1 ; lanes 16-31 -> v0=K2,v1=K3.
// B (4x16): v0 = K0 row (lanes 0-15) / K1 row (lanes 16-31), v1 = K2/K3.
// C/D 16x16 f32: VGPR r, lane l -> (M = r + 8*(l/16), N = l%16).
// ---------------------------------------------------------------------------
__global__ void engmf_gram(const float* __restrict__ X, float* __restrict__ part, int n)
{
    __shared__ float tile[TILE_F];
    __shared__ float lgram[8][36];
    __shared__ float lcs[8][12];

    const int lane = threadIdx.x & 31;
    const int wave = threadIdx.x >> 5;
    const int ln   = lane & 15;
    const bool hi  = lane >= 16;
    const int cl   = (ln < 6) ? ln : 0;        // clamped dim index (branchless)
    const float msk = (ln < 6) ? 1.f : 0.f;

    v8f acc0 = {};
    v8f acc1 = {};
    float cs = 0.f;

    const int ntiles = n / TILE_P;
    const unsigned ldsBase = (unsigned)(size_t)&tile[0];

    for (int tb = blockIdx.x; tb < ntiles; tb += gridDim.x) {
        // ---- async stage: 24KB tile, 6 x b128 per thread, coalesced ----
        const unsigned long long gtile =
            (unsigned long long)(size_t)X + (unsigned long long)tb * (TILE_F * 4ull);
#pragma unroll
        for (int j = 0; j < 6; ++j) {
            const unsigned off = (unsigned)(j * 4096 + threadIdx.x * 16);
            const unsigned ldsaddr = ldsBase + off;
            const unsigned long long gaddr = gtile + off;
            asm volatile("global_load_async_to_lds_b128 %0, %1, off"
                         :: "v"(ldsaddr), "v"(gaddr) : "memory");
        }
        asm volatile("s_wait_asynccnt 0x0" ::: "memory");
        __syncthreads();

        // ---- consume: 256 quads/tile, 32 per wave, 2 quads (2 WMMAs)/iter ----
        const int qbase = wave * 32;
#pragma unroll 2
        for (int qq = 0; qq < 32; qq += 2) {
            const int p0 = (qbase + qq) * 4;       // first particle of quad 0
            float xa[4], xb[4];
#pragma unroll
            for (int k = 0; k < 4; ++k) xa[k] = tile[(p0 + k) * 6 + cl] * msk;
#pragma unroll
            for (int k = 0; k < 4; ++k) xb[k] = tile[(p0 + 4 + k) * 6 + cl] * msk;

            v2f a0, b0, a1, b1;
            a0[0] = hi ? xa[2] : xa[0];  a0[1] = hi ? xa[3] : xa[1];
            b0[0] = hi ? xa[1] : xa[0];  b0[1] = hi ? xa[3] : xa[2];
            a1[0] = hi ? xb[2] : xb[0];  a1[1] = hi ? xb[3] : xb[1];
            b1[0] = hi ? xb[1] : xb[0];  b1[1] = hi ? xb[3] : xb[2];
            cs += a0[0] + a0[1] + a1[0] + a1[1];
            acc0 = __builtin_amdgcn_wmma_f32_16x16x4_f32(false, a0, false, b0,
                                                         (short)0, acc0, false, false);
            acc1 = __builtin_amdgcn_wmma_f32_16x16x4_f32(false, a1, false, b1,
                                                         (short)0, acc1, false, false);
        }
        __syncthreads();   // LDS reuse next tile
    }

    // ---- tail (block 0): branchless clamped global loads ----
    if (blockIdx.x == 0) {
        const int qstart = ntiles * (TILE_P / 4);
        const int nquads = (n + 3) >> 2;
        for (int q = qstart + wave; q < nquads; q += 8) {
            const int p0 = q << 2;
            float xq[4];
#pragma unroll
            for (int k = 0; k < 4; ++k) {
                const int p  = p0 + k;
                const int pc = (p < n) ? p : (n - 1);
                const float v = X[(size_t)pc * 6 + cl];
                xq[k] = (p < n) ? v * msk : 0.f;
            }
            v2f a, b;
            a[0] = hi ? xq[2] : xq[0];  a[1] = hi ? xq[3] : xq[1];
            b[0] = hi ? xq[1] : xq[0];  b[1] = hi ? xq[3] : xq[2];
            cs += a[0] + a[1];
            acc0 = __builtin_amdgcn_wmma_f32_16x16x4_f32(false, a, false, b,
                                                         (short)0, acc0, false, false);
        }
    }

    const v8f acc = acc0 + acc1;

    if (lane < 6) {
#pragma unroll
        for (int r = 0; r < 6; ++r) lgram[wave][r * 6 + lane] = acc[r];
    }
    if (ln < 6) lcs[wave][(hi ? 6 : 0) + ln] = cs;
    __syncthreads();

    const int t = threadIdx.x;
    if (t < 36) {
        float s = 0.f;
        for (int w = 0; w < 8; ++w) s += lgram[w][t];
        part[blockIdx.x * 42 + t] = s;
    } else if (t < 42) {
        const int d = t - 36;
        float s = 0.f;
        for (int w = 0; w < 8; ++w) s += lcs[w][d] + lcs[w][6 + d];
        part[blockIdx.x * 42 + t] = s;
    }
}

// Reduce block partials -> mean, emp_cov, localization, bandwidth -> P (6x6)
__global__ void engmf_finalize_P(const float* __restrict__ part, int nb,
                                 float* __restrict__ P, float bw, int n)
{
    __shared__ float gram[36];
    __shared__ float cs[6];
    const int t = threadIdx.x;
    if (t < 36) {
        float s = 0.f;
        for (int b = 0; b < nb; ++b) s += part[b * 42 + t];
        gram[t] = s;
    }
    if (t < 6) {
        float s = 0.f;
        for (int b = 0; b < nb; ++b) s += part[b * 42 + 36 + t];
        cs[t] = s;
    }
    __syncthreads();
    if (t < 36) {
        const int i = t / 6, j = t % 6;
        const float mi = cs[i] / (float)n;
        const float mj = cs[j] / (float)n;
        const float cov = (gram[t] - (float)n * mi * mj) / (float)(n - 1);
        const float dd = (float)(i - j);
        const float rho = expf(-dd * dd / (2.f * 3.f * 3.f));  // L_LOC = 3
        P[t] = bw * cov * rho;
    }
}

// ---------------------------------------------------------------------------
// Per-particle EnGMF update (shared by logw pass and the resampled pass).
// ---------------------------------------------------------------------------
__device__ inline float engmf_point(const float x[6], const float* __restrict__ H,
                                    const float* __restrict__ R, const float* __restrict__ y,
                                    const float P[36], float post[6], float pcov[36],
                                    bool need_cov)
{
    float t[6], g[6];
#pragma unroll
    for (int j = 0; j < 6; ++j) { float tj = tanhf(x[j]); t[j] = tj; g[j] = 1.f - tj * tj; }

    float J[3][6];
#pragma unroll
    for (int i = 0; i < 3; ++i)
#pragma unroll
        for (int j = 0; j < 6; ++j) J[i][j] = H[i * 6 + j] * g[j];

    float T[3][6];
#pragma unroll
    for (int i = 0; i < 3; ++i)
#pragma unroll
        for (int j = 0; j < 6; ++j) {
            float s = 0.f;
#pragma unroll
            for (int k = 0; k < 6; ++k) s += J[i][k] * P[k * 6 + j];
            T[i][j] = s;
        }

    float Sr[3][3];
#pragma unroll
    for (int i = 0; i < 3; ++i)
#pragma unroll
        for (int j = 0; j < 3; ++j) {
            float s = 0.f;
#pragma unroll
            for (int k = 0; k < 6; ++k) s += T[i][k] * J[j][k];
            Sr[i][j] = s + R[i * 3 + j];
        }
    const float S00 = Sr[0][0];
    const float S10 = 0.5f * (Sr[1][0] + Sr[0][1]);
    const float S20 = 0.5f * (Sr[2][0] + Sr[0][2]);
    const float S11 = Sr[1][1];
    const float S21 = 0.5f * (Sr[2][1] + Sr[1][2]);
    const float S22 = Sr[2][2];

    // 3x3 Cholesky
    const float L00 = sqrtf(fmaxf(S00, 1e-30f));
    const float L10 = S10 / L00;
    const float L20 = S20 / L00;
    const float L11 = sqrtf(fmaxf(S11 - L10 * L10, 1e-30f));
    const float L21 = (S21 - L20 * L10) / L11;
    const float L22 = sqrtf(fmaxf(S22 - L20 * L20 - L21 * L21, 1e-30f));

    // Kp = S^{-1} T  (3x6); gain K = Kp^T
    float Kp[3][6];
#pragma unroll
    for (int c = 0; c < 6; ++c) {
        const float u0 = T[0][c] / L00;
        const float u1 = (T[1][c] - L10 * u0) / L11;
        const float u2 = (T[2][c] - L20 * u0 - L21 * u1) / L22;
        const float w2 = u2 / L22;
        const float w1 = (u1 - L21 * w2) / L11;
        const float w0 = (u0 - L10 * w1 - L20 * w2) / L00;
        Kp[0][c] = w0; Kp[1][c] = w1; Kp[2][c] = w2;
    }

    float innov[3];
#pragma unroll
    for (int i = 0; i < 3; ++i) {
        float s = 0.f;
#pragma unroll
        for (int j = 0; j < 6; ++j) s += H[i * 6 + j] * t[j];
        innov[i] = y[i] - s;
    }

#pragma unroll
    for (int d = 0; d < 6; ++d)
        post[d] = x[d] + Kp[0][d] * innov[0] + Kp[1][d] * innov[1] + Kp[2][d] * innov[2];

    if (need_cov) {
#pragma unroll
        for (int a = 0; a < 6; ++a)
#pragma unroll
            for (int b = 0; b < 6; ++b)
                pcov[a * 6 + b] = P[a * 6 + b]
                                - (Kp[0][a] * T[0][b] + Kp[1][a] * T[1][b] + Kp[2][a] * T[2][b]);
    }

    const float z0 = innov[0] / L00;
    const float z1 = (innov[1] - L10 * z0) / L11;
    const float z2 = (innov[2] - L20 * z0 - L21 * z1) / L22;
    return -0.5f * (z0 * z0 + z1 * z1 + z2 * z2)
           - (logf(L00) + logf(L11) + logf(L22))
           - 0.5f * 3.f * LOG2PI_F;
}

__global__ void engmf_logw(const float* __restrict__ X, const float* __restrict__ H,
                           const float* __restrict__ R, const float* __restrict__ y,
                           const float* __restrict__ P, float* __restrict__ logw, int n)
{
    const int i = blockIdx.x * blockDim.x + threadIdx.x;
    if (i >= n) return;
    float Pl[36];
#pragma unroll
    for (int k = 0; k < 36; ++k) Pl[k] = P[k];
    // 24B rows are always 8B aligned -> b64 loads
    const float2* __restrict__ xp = (const float2*)(X + (size_t)i * 6);
    float x[6];
#pragma unroll
    for (int h = 0; h < 3; ++h) { float2 v = xp[h]; x[2 * h] = v.x; x[2 * h + 1] = v.y; }
    float post[6];
    logw[i] = engmf_point(x, H, R, y, Pl, post, nullptr, false);
}

// ---------------------------------------------------------------------------
// Softmax: deterministic two-pass (max, sum-exp) tree reductions.
// ---------------------------------------------------------------------------
__global__ void engmf_red_max(const float* __restrict__ v, float* __restrict__ part, int n)
{
    __shared__ float s[256];
    float m = -3.4e38f;
    for (int i = blockIdx.x * blockDim.x + threadIdx.x; i < n; i += gridDim.x * blockDim.x)
        m = fmaxf(m, v[i]);
    s[threadIdx.x] = m;
    __syncthreads();
    for (int o = 128; o > 0; o >>= 1) {
        if (threadIdx.x < o) s[threadIdx.x] = fmaxf(s[threadIdx.x], s[threadIdx.x + o]);
        __syncthreads();
    }
    if (threadIdx.x == 0) part[blockIdx.x] = s[0];
}

__global__ void engmf_final_max(const float* __restrict__ part, int nb, float* __restrict__ scal)
{
    __shared__ float s[512];
    float m = -3.4e38f;
    for (int i = threadIdx.x; i < nb; i += blockDim.x) m = fmaxf(m, part[i]);
    s[threadIdx.x] = m;
    __syncthreads();
    for (int o = 256; o > 0; o >>= 1) {
        if (threadIdx.x < o) s[threadIdx.x] = fmaxf(s[threadIdx.x], s[threadIdx.x + o]);
        __syncthreads();
    }
    if (threadIdx.x == 0) scal[0] = s[0];
}

__global__ void engmf_red_sumexp(const float* __restrict__ v, const float* __restrict__ scal,
                                 float* __restrict__ part, int n)
{
    __shared__ float s[256];
    const float mx = scal[0];
    float a = 0.f;
    for (int i = blockIdx.x * blockDim.x + threadIdx.x; i < n; i += gridDim.x * blockDim.x)
        a += expf(v[i] - mx);
    s[threadIdx.x] = a;
    __syncthreads();
    for (int o = 128; o > 0; o >>= 1) {
        if (threadIdx.x < o) s[threadIdx.x] += s[threadIdx.x + o];
        __syncthreads();
    }
    if (threadIdx.x == 0) part[blockIdx.x] = s[0];
}

__global__ void engmf_final_sum(const float* __restrict__ part, int nb, float* __restrict__ scal)
{
    __shared__ float s[512];
    float a = 0.f;
    for (int i = threadIdx.x; i < nb; i += blockDim.x) a += part[i];
    s[threadIdx.x] = a;
    __syncthreads();
    for (int o = 256; o > 0; o >>= 1) {
        if (threadIdx.x < o) s[threadIdx.x] += s[threadIdx.x + o];
        __syncthreads();
    }
    if (threadIdx.x == 0) scal[1] = s[0];
}

__global__ void engmf_weights(const float* __restrict__ logw, const float* __restrict__ scal,
                              float* __restrict__ w, int n)
{
    const int i = blockIdx.x * blockDim.x + threadIdx.x;
    if (i >= n) return;
    w[i] = expf(logw[i] - scal[0]) / scal[1];
}

// ---------------------------------------------------------------------------
// Systematic resampling: chunk sums -> single-block scan -> binary search.
// ---------------------------------------------------------------------------
__global__ void engmf_chunk_sums(const float* __restrict__ w, float* __restrict__ chs, int n)
{
    __shared__ float s[256];
    const int base = blockIdx.x * CHUNK;
    float a = 0.f;
    for (int k = threadIdx.x; k < CHUNK; k += blockDim.x) {
        const int i = base + k;
        if (i < n) a += w[i];
    }
    s[threadIdx.x] = a;
    __syncthreads();
    for (int o = 128; o > 0; o >>= 1) {
        if (threadIdx.x < o) s[threadIdx.x] += s[threadIdx.x + o];
        __syncthreads();
    }
    if (threadIdx.x == 0) chs[blockIdx.x] = s[0];
}

__global__ void engmf_scan(const float* __restrict__ chs, float* __restrict__ ccdf, int nch)
{
    __shared__ float s[1024];
    const int t = threadIdx.x;
    s[t] = (t < nch) ? chs[t] : 0.f;
    __syncthreads();
    for (int o = 1; o < 1024; o <<= 1) {
        const float add = (t >= o) ? s[t - o] : 0.f;
        __syncthreads();
        s[t] += add;
        __syncthreads();
    }
    if (t < nch) ccdf[t] = s[t];
}

__global__ void engmf_select(const float* __restrict__ w, const float* __restrict__ ccdf,
                             int nch, int* __restrict__ idx, int n)
{
    const int i = blockIdx.x * blockDim.x + threadIdx.x;
    if (i >= n) return;
    const float total = ccdf[nch - 1];
    const float u = ((float)i + 0.5f) / (float)n * total;
    int lo = 0, hi = nch - 1;
    while (lo < hi) {
        const int mid = (lo + hi) >> 1;
        if (ccdf[mid] < u) lo = mid + 1; else hi = mid;
    }
    float acc = (lo > 0) ? ccdf[lo - 1] : 0.f;
    const int start = lo * CHUNK;
    const int end = min(start + CHUNK, n);
    int j = end - 1;
    for (int k = start; k < end; ++k) {
        acc += w[k];
        if (acc >= u) { j = k; break; }
    }
    idx[i] = j;
}

// ---------------------------------------------------------------------------
// Resampled pass: recompute update at idx[i] (compute is free, bytes are not:
// avoids a 72MB covariance round-trip), chol6 + Gaussian sample.
// ---------------------------------------------------------------------------
__device__ inline unsigned engmf_hash(unsigned v)
{
    v = v * 747796405u + 2891336453u;
    unsigned w = ((v >> ((v >> 28) + 4u)) ^ v) * 277803737u;
    return (w >> 22) ^ w;
}
__device__ inline float engmf_u01(unsigned h) { return ((float)h + 0.5f) * 2.3283064365386963e-10f; }

__global__ void engmf_sample(const float* __restrict__ X, const float* __restrict__ H,
                             const float* __restrict__ R, const float* __restrict__ y,
                             const float* __restrict__ P, const int* __restrict__ idx,
                             float* __restrict__ out_s, float* __restrict__ out_c, int n)
{
    const int i = blockIdx.x * blockDim.x + threadIdx.x;
    if (i >= n) return;
    float Pl[36];
#pragma unroll
    for (int k = 0; k < 36; ++k) Pl[k] = P[k];

    const int j = idx[i];
    const float2* __restrict__ xp = (const float2*)(X + (size_t)j * 6);
    float x[6];
#pragma unroll
    for (int h = 0; h < 3; ++h) { float2 v = xp[h]; x[2 * h] = v.x; x[2 * h + 1] = v.y; }

    float post[6], pcov[36];
    (void)engmf_point(x, H, R, y, Pl, post, pcov, true);

    // 144B rows always 8B aligned -> b64 stores
    float2* __restrict__ cp = (float2*)(out_c + (size_t)i * 36);
#pragma unroll
    for (int k = 0; k < 18; ++k) { float2 v; v.x = pcov[2 * k]; v.y = pcov[2 * k + 1]; cp[k] = v; }

    // 6x6 Cholesky of posterior covariance
    float Lc[6][6];
#pragma unroll
    for (int a = 0; a < 6; ++a)
#pragma unroll
        for (int b = 0; b < 6; ++b) Lc[a][b] = 0.f;
#pragma unroll
    for (int a = 0; a < 6; ++a) {
#pragma unroll
        for (int b = 0; b < 6; ++b) {
            if (b > a) continue;
            float s = pcov[a * 6 + b];
#pragma unroll
            for (int k = 0; k < 6; ++k)
                if (k < b) s -= Lc[a][k] * Lc[b][k];
            if (a == b) Lc[a][a] = sqrtf(fmaxf(s, 1e-20f));
            else        Lc[a][b] = s / Lc[b][b];
        }
    }

    // deterministic counter-hash Box-Muller normals
    float z[6];
#pragma unroll
    for (int k = 0; k < 3; ++k) {
        const unsigned base = (unsigned)i * 6u + (unsigned)(2 * k);
        const float u1 = engmf_u01(engmf_hash(base ^ 0x9E3779B9u));
        const float u2 = engmf_u01(engmf_hash(base + 1u));
        const float r  = sqrtf(-2.f * logf(fmaxf(u1, 1e-12f)));
        const float th = 6.283185307179586f * u2;
        z[2 * k]     = r * cosf(th);
        z[2 * k + 1] = r * sinf(th);
    }

    float smp[6];
#pragma unroll
    for (int d = 0; d < 6; ++d) {
        float s = post[d];
#pragma unroll
        for (int k = 0; k < 6; ++k) s += Lc[d][k] * z[k];
        smp[d] = s;
    }
    float2* __restrict__ sp = (float2*)(out_s + (size_t)i * 6);
#pragma unroll
    for (int h = 0; h < 3; ++h) { float2 v; v.x = smp[2 * h]; v.y = smp[2 * h + 1]; sp[h] = v; }
}

// ---------------------------------------------------------------------------
extern "C" void kernel_launch(void* const* d_in, const int* in_sizes, int n_in,
                              void* d_out, int out_size, void* d_ws, size_t ws_size,
                              hipStream_t stream)
{
    const float* X = (const float*)d_in[0];   // [n,6]
    const float* y = (const float*)d_in[1];   // [3]
    const float* H = (const float*)d_in[2];   // [3,6]
    const float* R = (const float*)d_in[3];   // [3,3]
    const int n = in_sizes[0] / 6;

    float* out      = (float*)d_out;
    float* out_smp  = out;                        // n*6
    float* out_cov  = out + (size_t)n * 6;        // n*36
    float* out_wts  = out + (size_t)n * 42;       // n

    // workspace layout (floats)
    float* wsf      = (float*)d_ws;
    float* ws_P     = wsf;                         // 36 (pad to 64)
    float* ws_logw  = wsf + 64;                    // n
    float* ws_part  = ws_logw + n;                 // NB_GRAM*42
    float* ws_maxp  = ws_part + NB_GRAM * 42;      // NB_RED
    float* ws_sump  = ws_maxp + NB_RED;            // NB_RED
    float* ws_scal  = ws_sump + NB_RED;            // 8
    float* ws_chs   = ws_scal + 8;                 // 1024
    float* ws_ccdf  = ws_chs + 1024;               // 1024
    int*   ws_idx   = (int*)(ws_ccdf + 1024);      // n

    const int nch = (n + CHUNK - 1) / CHUNK;       // 977 for n=500000 (<=1024)
    const int gpp = (n + 255) / 256;               // per-particle grid

    // Silverman bandwidth (host-side deterministic scalar)
    const double bw = pow(4.0 / ((double)n * (6.0 + 2.0)), 2.0 / (6.0 + 4.0));

    engmf_gram<<<NB_GRAM, 256, 0, stream>>>(X, ws_part, n);
    engmf_finalize_P<<<1, 64, 0, stream>>>(ws_part, NB_GRAM, ws_P, (float)bw, n);
    engmf_logw<<<gpp, 256, 0, stream>>>(X, H, R, y, ws_P, ws_logw, n);
    engmf_red_max<<<NB_RED, 256, 0, stream>>>(ws_logw, ws_maxp, n);
    engmf_final_max<<<1, 512, 0, stream>>>(ws_maxp, NB_RED, ws_scal);
    engmf_red_sumexp<<<NB_RED, 256, 0, stream>>>(ws_logw, ws_scal, ws_sump, n);
    engmf_final_sum<<<1, 512, 0, stream>>>(ws_sump, NB_RED, ws_scal);
    engmf_weights<<<gpp, 256, 0, stream>>>(ws_logw, ws_scal, out_wts, n);
    engmf_chunk_sums<<<nch, 256, 0, stream>>>(out_wts, ws_chs, n);
    engmf_scan<<<1, 1024, 0, stream>>>(ws_chs, ws_ccdf, nch);
    engmf_select<<<gpp, 256, 0, stream>>>(out_wts, ws_ccdf, nch, ws_idx, n);
    engmf_sample<<<gpp, 256, 0, stream>>>(X, H, R, y, ws_P, ws_idx, out_smp, out_cov, n);
}